// SGC_29386166239464
// MI455X (gfx1250) — compile-verified
//
#include <hip/hip_runtime.h>
#include <hip/hip_bf16.h>
#include <math.h>

typedef __attribute__((ext_vector_type(2))) float v2f;
typedef __attribute__((ext_vector_type(8))) float v8f;

#define SGC_D 512
#define SGC_C 40

// ---------------- degree / normalization ----------------

__global__ void sgc_deg_init(float* __restrict__ deg, int N) {
    int i = blockIdx.x * blockDim.x + threadIdx.x;
    if (i < N) deg[i] = 1.0f;  // self-loop contributes exactly 1 per node
}

__global__ void sgc_deg_accum(const int* __restrict__ dst, float* __restrict__ deg, int E) {
    int e = blockIdx.x * blockDim.x + threadIdx.x;
    if (e < E) atomicAdd(&deg[dst[e]], 1.0f);
}

__global__ void sgc_dinv(float* __restrict__ deg, int N) {
    int i = blockIdx.x * blockDim.x + threadIdx.x;
    if (i < N) {
        float d = deg[i];
        deg[i] = (d > 0.0f) ? rsqrtf(d) : 0.0f;
    }
}

// ---------------- propagation (one hop) ----------------
// Self-loop term: hout[i] = dinv[i]^2 * hin[i]   (also serves as the zero-init)
__global__ void sgc_self_init(const float* __restrict__ hin, float* __restrict__ hout,
                              const float* __restrict__ dinv, int N) {
    long long t = (long long)blockIdx.x * blockDim.x + threadIdx.x;
    int node = (int)(t >> 7);   // 128 float4 segments per node (512 floats)
    int seg  = (int)(t & 127);
    if (node >= N) return;
    float w = dinv[node];
    w *= w;
    const float4* ip = (const float4*)(hin + (size_t)node * SGC_D);
    float4* op = (float4*)(hout + (size_t)node * SGC_D);
    float4 v = ip[seg];
    op[seg] = make_float4(v.x * w, v.y * w, v.z * w, v.w * w);
}

// Edge scatter: hout[dst] += dinv[src]*dinv[dst] * hin[src]; 128 lanes/edge, float4 each.
__global__ void sgc_edge_prop(const float* __restrict__ hin, float* __restrict__ hout,
                              const int* __restrict__ src, const int* __restrict__ dst,
                              const float* __restrict__ dinv, int E) {
    long long t = (long long)blockIdx.x * blockDim.x + threadIdx.x;
    int e   = (int)(t >> 7);
    int seg = (int)(t & 127);
    if (e >= E) return;
    int s = src[e];
    int d = dst[e];
    float nrm = dinv[s] * dinv[d];
    const float4* ip = (const float4*)(hin + (size_t)s * SGC_D);
    float4 v = ip[seg];
    float* op = hout + (size_t)d * SGC_D + (size_t)seg * 4;
    atomicAdd(op + 0, nrm * v.x);
    atomicAdd(op + 1, nrm * v.y);
    atomicAdd(op + 2, nrm * v.z);
    atomicAdd(op + 3, nrm * v.w);
}

// ---------------- WMMA GEMM (fp32) + row-wise log_softmax ----------------
// One wave computes a 16-row x 48-col logit tile (3 N-tiles of 16, cols>=40 masked)
// via V_WMMA_F32_16X16X4_F32, then log-softmax per row with width-16 shfl_xor.
//
// A (16x4, fp32): lane<16 -> M=lane, {K=0,K=1}; lane>=16 -> M=lane-16, {K=2,K=3}
// B (4x16, fp32): lane<16 -> N=lane, {K=0,K=1}; lane>=16 -> N=lane-16, {K=2,K=3}
// C/D (16x16):    VGPR v: lane<16 -> M=v, N=lane; lane>=16 -> M=v+8, N=lane-16
//
// OOB handling is done entirely OUTSIDE the k-loop:
//  * rows >= N: pointer clamped to row 0; the resulting garbage D-rows are
//    never stored and never mix into the per-row softmax reductions.
//  * cols >= C: pointer clamped to W row 0; the garbage logits are replaced
//    with -inf in the epilogue before the lane reduction, so they vanish.
// => inner loop is 4 unconditional 8B loads + 3 WMMAs, no exec masking.
__global__ void __launch_bounds__(128)
sgc_gemm_lsm(const float* __restrict__ h, const float* __restrict__ Wm,
             const float* __restrict__ bias, float* __restrict__ out, int N) {
    const int wave = threadIdx.x >> 5;
    const int lane = threadIdx.x & 31;
    const int rowbase = (blockIdx.x * 4 + wave) * 16;
    if (rowbase >= N) return;

    const int half = lane >> 4;   // 0: K=0..1 / rows M=v ; 1: K=2..3 / rows M=v+8
    const int l    = lane & 15;

    const int rowA = rowbase + l;
    const float* __restrict__ arow = h + (size_t)(rowA < N ? rowA : 0) * SGC_D + half * 2;

    const int c0 = l, c1 = 16 + l, c2 = 32 + l;
    const bool c2ok = (c2 < SGC_C);
    const float* __restrict__ w0 = Wm + (size_t)c0 * SGC_D + half * 2;
    const float* __restrict__ w1 = Wm + (size_t)c1 * SGC_D + half * 2;
    const float* __restrict__ w2 = Wm + (size_t)(c2ok ? c2 : 0) * SGC_D + half * 2;

    v8f acc0 = {}, acc1 = {}, acc2 = {};
#pragma unroll 8
    for (int k = 0; k < SGC_D; k += 4) {
        v2f a  = *(const v2f*)(arow + k);
        v2f b0 = *(const v2f*)(w0 + k);
        v2f b1 = *(const v2f*)(w1 + k);
        v2f b2 = *(const v2f*)(w2 + k);
        acc0 = __builtin_amdgcn_wmma_f32_16x16x4_f32(false, a, false, b0, (short)0, acc0, false, false);
        acc1 = __builtin_amdgcn_wmma_f32_16x16x4_f32(false, a, false, b1, (short)0, acc1, false, false);
        acc2 = __builtin_amdgcn_wmma_f32_16x16x4_f32(false, a, false, b2, (short)0, acc2, false, false);
    }

    const float bb0 = bias[c0];
    const float bb1 = bias[c1];
    const float bb2 = c2ok ? bias[c2] : 0.0f;

    float o0[8], o1[8], o2[8];
    for (int v = 0; v < 8; ++v) {
        float e0 = acc0[v] + bb0;
        float e1 = acc1[v] + bb1;
        float e2 = c2ok ? (acc2[v] + bb2) : -INFINITY;
        // row max over the 16 lanes of this half-wave (masks 1..8 stay in-half)
        float m = fmaxf(e0, fmaxf(e1, e2));
        for (int sh = 1; sh < 16; sh <<= 1)
            m = fmaxf(m, __shfl_xor(m, sh, 32));
        float s = __expf(e0 - m) + __expf(e1 - m) + (c2ok ? __expf(e2 - m) : 0.0f);
        for (int sh = 1; sh < 16; sh <<= 1)
            s += __shfl_xor(s, sh, 32);
        float L = m + __logf(s);
        o0[v] = e0 - L;
        o1[v] = e1 - L;
        o2[v] = e2 - L;
    }
    for (int v = 0; v < 8; ++v) {
        int row = rowbase + v + half * 8;
        if (row < N) {
            float* orow = out + (size_t)row * SGC_C;
            orow[c0] = o0[v];
            orow[c1] = o1[v];
            if (c2ok) orow[c2] = o2[v];
        }
    }
}

// ---------------- launch ----------------

extern "C" void kernel_launch(void* const* d_in, const int* in_sizes, int n_in,
                              void* d_out, int out_size, void* d_ws, size_t ws_size,
                              hipStream_t stream) {
    const float* x  = (const float*)d_in[0];
    const int*   ei = (const int*)d_in[1];
    const float* Wm = (const float*)d_in[2];
    const float* bb = (const float*)d_in[3];
    const int D = SGC_D;
    const int N = in_sizes[0] / D;
    const int E = in_sizes[1] / 2;
    const int* src = ei;
    const int* dst = ei + E;
    float* out = (float*)d_out;

    // workspace layout: dinv[N] | h1[N*D] | h2[N*D]
    float* dinv = (float*)d_ws;
    float* h1 = dinv + N;              // byte offset N*4 = 400000 (16B aligned)
    float* h2 = h1 + (size_t)N * D;

    // 1) degree (self-loop baked in as 1.0) and dinv
    sgc_deg_init<<<(N + 255) / 256, 256, 0, stream>>>(dinv, N);
    sgc_deg_accum<<<(E + 255) / 256, 256, 0, stream>>>(dst, dinv, E);
    sgc_dinv<<<(N + 255) / 256, 256, 0, stream>>>(dinv, N);

    const long long nodeThreads = (long long)N * 128;
    const long long edgeThreads = (long long)E * 128;
    const unsigned nodeBlocks = (unsigned)((nodeThreads + 255) / 256);
    const unsigned edgeBlocks = (unsigned)((edgeThreads + 255) / 256);

    // 2) hop 1: x -> h1
    sgc_self_init<<<nodeBlocks, 256, 0, stream>>>(x, h1, dinv, N);
    sgc_edge_prop<<<edgeBlocks, 256, 0, stream>>>(x, h1, src, dst, dinv, E);

    // 3) hop 2: h1 -> h2
    sgc_self_init<<<nodeBlocks, 256, 0, stream>>>(h1, h2, dinv, N);
    sgc_edge_prop<<<edgeBlocks, 256, 0, stream>>>(h1, h2, src, dst, dinv, E);

    // 4) WMMA GEMM + log_softmax: h2 [N,512] x W^T [512,40] + b -> out [N,40]
    const unsigned gemmBlocks = (unsigned)((N + 63) / 64);  // 4 waves x 16 rows per block
    sgc_gemm_lsm<<<gemmBlocks, 128, 0, stream>>>(h2, Wm, bb, out, N);
}